// QuantumConvLayer_84421877170200
// MI455X (gfx1250) — compile-verified
//
#include <hip/hip_runtime.h>

typedef __attribute__((ext_vector_type(2))) float v2f;
typedef __attribute__((ext_vector_type(8))) float v8f;

// x:   [B, 1, 1024, 1024] f32
// out: [B, 4,  512,  512] f32
// One wave per (b, output-row i). 16 iterations x 32 patches per wave.
// Patch group g (16 patches) is one V_WMMA_F32_16X16X4_F32:
//   A (16x4) = U replicated (A[m][k] = U[m&3][k])   -> per-lane v2f constants
//   B (4x16) = normalized psi columns               -> per-lane v2f from 8B loads
//   D (16x16): VGPR k, lanes 0-15 = amp[k][patch]; lanes 16-31 duplicate.
__global__ __launch_bounds__(256) void quanv_wmma_kernel(
    const float* __restrict__ x, float* __restrict__ out) {
  const int lane = threadIdx.x & 31;
  const int wid  = (blockIdx.x << 3) + (threadIdx.x >> 5);  // global wave id
  const int b    = wid >> 9;                                // batch
  const int i    = wid & 511;                               // output row

  const int m    = lane & 3;   // U row (replicated)
  const int half = lane >> 4;  // 0: holds K=0,1   1: holds K=2,3
  const int p    = lane & 15;  // patch-in-group / B column

  // U = CX * (H kron H), entries are +-0.5:
  //   row0: + + + +   row1: + - + -   row2: + - - +   row3: + + - -
  const float s1 = ((m ^ (m >> 1)) & 1) ? -0.5f : 0.5f;  // column 1 sign
  const float s2 = (m & 2) ? -0.5f : 0.5f;               // column 2 sign
  const float s3 = (m & 1) ? -0.5f : 0.5f;               // column 3 sign
  v2f a;
  a.x = half ? s2 : 0.5f;   // K = 2*half
  a.y = half ? s3 : s1;     // K = 2*half + 1

  // depolarizing mixture constants (match reference's double->f32 rounding)
  const float noise = (float)(1.0 - (1.0 - 0.001) * (1.0 - 0.001) * (1.0 - 0.01));
  const float w1 = 1.0f - noise;
  const float w0 = noise * 0.25f;

  // input row for this half-wave: row 2i (lanes 0-15) or 2i+1 (lanes 16-31)
  const float* rowp = x + ((size_t)b << 20) + ((size_t)((i << 1) + half) << 10);
  // output base for (b, i); channel stride = 512*512 = 1<<18
  float* ob = out + ((size_t)b << 20) + ((size_t)i << 9);

#pragma unroll 2
  for (int c = 0; c < 16; ++c) {
    const int j0  = c << 5;              // first patch column of this chunk
    const int col = (j0 << 1) + (p << 1);

    // 8B per lane, 128B contiguous per half-wave, streaming (NT) hint
    v2f r1 = __builtin_nontemporal_load((const v2f*)(rowp + col));        // patches j0..j0+15
    v2f r2 = __builtin_nontemporal_load((const v2f*)(rowp + col + 32));   // patches j0+16..j0+31

    // amplitude-encoding norm: cross-half reduction of |patch|^2
    float q1 = r1.x * r1.x + r1.y * r1.y;
    float q2 = r2.x * r2.x + r2.y * r2.y;
    float n1 = q1 + __shfl_xor(q1, 16, 32) + 1e-12f;
    float n2 = q2 + __shfl_xor(q2, 16, 32) + 1e-12f;
    float rn1 = rsqrtf(n1);
    float rn2 = rsqrtf(n2);
    v2f b1; b1.x = r1.x * rn1; b1.y = r1.y * rn1;
    v2f b2; b2.x = r2.x * rn2; b2.y = r2.y * rn2;

    // amp = U * psi, 16 patches per WMMA
    v8f z = {};
    v8f d1 = __builtin_amdgcn_wmma_f32_16x16x4_f32(
        false, a, false, b1, (short)0, z, false, false);
    v8f d2 = __builtin_amdgcn_wmma_f32_16x16x4_f32(
        false, a, false, b2, (short)0, z, false, false);

    // probs + noise, fully coalesced 128B store per channel:
    // lanes 0-15 take group-1 columns, lanes 16-31 take group-2 duplicates.
    float* o = ob + j0 + lane;
#pragma unroll
    for (int k = 0; k < 4; ++k) {
      float amp = (half == 0) ? d1[k] : d2[k];
      __builtin_nontemporal_store(fmaf(w1 * amp, amp, w0),
                                  o + ((size_t)k << 18));
    }
  }
}

extern "C" void kernel_launch(void* const* d_in, const int* in_sizes, int n_in,
                              void* d_out, int out_size, void* d_ws, size_t ws_size,
                              hipStream_t stream) {
  const float* x = (const float*)d_in[0];
  float* out = (float*)d_out;

  const int B = in_sizes[0] >> 20;      // elements / (1024*1024)
  const int waves = B * 512;            // one wave per (b, row)
  const int blocks = waves >> 3;        // 8 waves (256 thr) per block

  hipLaunchKernelGGL(quanv_wmma_kernel, dim3(blocks), dim3(256), 0, stream,
                     x, out);
}